// CRF_15599321219276
// MI455X (gfx1250) — compile-verified
//
#include <hip/hip_runtime.h>
#include <hip/hip_bf16.h>

// ---------------------------------------------------------------------------
// CRF loss for MI455X (gfx1250).
//
// Forward recursion rewritten as a GEMM:
//   fv_new[n] = emit[n] + m + log( (exp(trans) @ exp(fv - m))[n] )
// exp(trans) is a fixed 128x128 f32 matrix; each workgroup advances 16
// sentences, so each step is a 128x128x16 f32 GEMM via V_WMMA_F32_16X16X4_F32
// (wave32, 8 waves = 8 row-tiles of 16). The A operand (exp(trans)) is
// loop-invariant -> hoisted into 64 VGPRs per lane. Per step, all 32 B
// operands are preloaded to registers (one ds burst), then the 32 WMMAs run
// in a 4-chain round-robin with no interleaved LDS waits on the sequential-L
// critical path.
// ---------------------------------------------------------------------------

#define BB 512
#define LL 512
#define TT 128
#define NB 16          // sentences per workgroup
#define ROWP 132       // padded LDS row (kills 512B-stride bank conflicts)
#define START_TAG 126
#define STOP_TAG 127
#define NEGV (-10000.0f)

typedef __attribute__((ext_vector_type(2))) float v2f;
typedef __attribute__((ext_vector_type(8))) float v8f;

__global__ void __launch_bounds__(256, 1)
crf_forward(const float* __restrict__ feats, const float* __restrict__ trans,
            const int* __restrict__ lens, float* __restrict__ fwd_out)
{
    __shared__ float fvb[NB * ROWP];   // fv[b][t]
    __shared__ float xbT[NB * ROWP];   // exp(fv[b][t] - m[b])
    __shared__ float wmax[8 * NB];     // per-wave column max
    __shared__ float mbuf[NB];         // per-step shift m[b]

    const int tid = threadIdx.x;
    const int l   = tid & 31;              // lane in wave32
    const int w   = tid >> 5;              // wave 0..7 -> rows [16w,16w+16)
    const int hi  = l >> 4;                // half-wave
    const int lb  = l & 15;                // lane-in-half == N (batch) index
    const int bbase = blockIdx.x * NB;

    // ---- hoist A = exp(trans) tile rows into registers (loop-invariant) ----
    // ISA 32-bit A 16x4 layout: lanes 0-15 hold K={0,1}, lanes 16-31 K={2,3}.
    // Per lane: row 16w+lb, K elements {4j+2hi, 4j+2hi+1} for j = 0..31.
    const float* trow = trans + (16 * w + lb) * TT + 2 * hi;
    v2f areg[32];
    #pragma unroll
    for (int j = 0; j < 32; ++j) {
        float2 t2 = *(const float2*)(trow + 4 * j);
        v2f a;
        a.x = __expf(t2.x);                // exp(-10000) == 0.0f
        a.y = __expf(t2.y);
        areg[j] = a;
    }

    // ---- init fv / wmax ----
    for (int idx = tid; idx < NB * TT; idx += 256) {
        int b = idx >> 7, t = idx & 127;
        fvb[b * ROWP + t] = (t == START_TAG) ? 0.0f : NEGV;
    }
    if (tid < 8 * NB) wmax[tid] = ((tid >> 4) == 7) ? 0.0f : NEGV; // START row lives in wave 7
    __syncthreads();

    const int    mylen  = lens[bbase + lb];
    const float* featb  = feats + (size_t)(bbase + lb) * LL * TT;
    const int    myrow0 = 16 * w + 8 * hi; // 8 contiguous output rows per lane (C-layout)

    const float* brow = xbT + lb * ROWP + 2 * hi;
    float* frow       = fvb + lb * ROWP + myrow0;

    for (int i = 0; i < LL; ++i) {
        // --- prefetch emit scores early (hidden under max/exp/WMMA) ---
        const float4* ep = (const float4*)(featb + (size_t)i * TT + myrow0);
        float4 e0 = ep[0];
        float4 e1 = ep[1];

        // --- stage 2: per-column shift m[b] and x = exp(fv - m) ---
        {
            int b = tid & 15, seg = tid >> 4;
            float m = wmax[b];
            #pragma unroll
            for (int ww = 1; ww < 8; ++ww) m = fmaxf(m, wmax[ww * 16 + b]);
            if (seg == 0) mbuf[b] = m;
            const float* fr = fvb + b * ROWP + seg * 8;
            float*       xr = xbT + b * ROWP + seg * 8;
            #pragma unroll
            for (int t = 0; t < 8; ++t) xr[t] = __expf(fr[t] - m);
        }
        __syncthreads();

        // --- stage 4a: burst-load all 32 B operands into registers ---
        v2f breg[32];
        #pragma unroll
        for (int j = 0; j < 32; ++j)
            breg[j] = *(const v2f*)(brow + 4 * j);

        // --- stage 4b: 32 WMMAs, 4 independent accumulator chains ---
        v8f acc0 = {}, acc1 = {}, acc2 = {}, acc3 = {};
        #pragma unroll
        for (int j = 0; j < 8; ++j) {
            acc0 = __builtin_amdgcn_wmma_f32_16x16x4_f32(
                       false, areg[4 * j + 0], false, breg[4 * j + 0], (short)0, acc0, false, false);
            acc1 = __builtin_amdgcn_wmma_f32_16x16x4_f32(
                       false, areg[4 * j + 1], false, breg[4 * j + 1], (short)0, acc1, false, false);
            acc2 = __builtin_amdgcn_wmma_f32_16x16x4_f32(
                       false, areg[4 * j + 2], false, breg[4 * j + 2], (short)0, acc2, false, false);
            acc3 = __builtin_amdgcn_wmma_f32_16x16x4_f32(
                       false, areg[4 * j + 3], false, breg[4 * j + 3], (short)0, acc3, false, false);
        }

        // --- stage 5: fv_new = emit + m + log(y); freeze past lens[b] ---
        {
            float m   = mbuf[lb];
            bool  upd = (i < mylen);
            float e[8] = {e0.x, e0.y, e0.z, e0.w, e1.x, e1.y, e1.z, e1.w};
            float mx = -3.0e38f;
            #pragma unroll
            for (int r = 0; r < 8; ++r) {
                float y   = (acc0[r] + acc1[r]) + (acc2[r] + acc3[r]);
                float nv  = e[r] + m + __logf(y);
                float old = frow[r];
                float sel = upd ? nv : old;
                frow[r] = sel;
                mx = fmaxf(mx, sel);
            }
            mx = fmaxf(mx, __shfl_xor(mx, 16, 32)); // combine both half-wave row groups
            if (hi == 0) wmax[w * 16 + lb] = mx;
        }
        __syncthreads();
    }

    // ---- terminal: forward_score[b] = logsumexp_t(fv[b][t] + trans[START][t]) ----
    if (tid < NB) {
        int b = tid;
        const float* srow = trans + START_TAG * TT;
        float m = -3.0e38f;
        for (int t = 0; t < TT; ++t) m = fmaxf(m, fvb[b * ROWP + t] + srow[t]);
        float s = 0.0f;
        for (int t = 0; t < TT; ++t) s += __expf(fvb[b * ROWP + t] + srow[t] - m);
        fwd_out[bbase + b] = m + __logf(s);
    }
}

__global__ void __launch_bounds__(256, 1)
crf_gold(const float* __restrict__ feats, const float* __restrict__ trans,
         const int* __restrict__ tags, const int* __restrict__ lens,
         float* __restrict__ gold_out)
{
    __shared__ float red[256];
    const int b   = blockIdx.x;
    const int len = lens[b];
    const int* tg = tags + (size_t)b * LL;

    float s = 0.0f;
    for (int j = threadIdx.x; j <= LL; j += 256) {
        int ps = (j == 0) ? START_TAG : tg[j - 1];
        int pt = (j < LL) ? tg[j] : STOP_TAG;
        if (j >= len) pt = STOP_TAG;
        if (j <= len) s += trans[pt * TT + ps];
        if (j < LL && j < len) s += feats[((size_t)b * LL + j) * TT + tg[j]];
    }
    red[threadIdx.x] = s;
    __syncthreads();
    for (int off = 128; off > 0; off >>= 1) {
        if (threadIdx.x < off) red[threadIdx.x] += red[threadIdx.x + off];
        __syncthreads();
    }
    if (threadIdx.x == 0) gold_out[b] = red[0];
}

__global__ void __launch_bounds__(256, 1)
crf_final(const float* __restrict__ fwd, const float* __restrict__ gold,
          float* __restrict__ out)
{
    __shared__ float red[256];
    float s = 0.0f;
    for (int i = threadIdx.x; i < BB; i += 256) s += fwd[i] - gold[i];
    red[threadIdx.x] = s;
    __syncthreads();
    for (int off = 128; off > 0; off >>= 1) {
        if (threadIdx.x < off) red[threadIdx.x] += red[threadIdx.x + off];
        __syncthreads();
    }
    if (threadIdx.x == 0) out[0] = red[0] / (float)BB;
}

extern "C" void kernel_launch(void* const* d_in, const int* in_sizes, int n_in,
                              void* d_out, int out_size, void* d_ws, size_t ws_size,
                              hipStream_t stream)
{
    const float* feats = (const float*)d_in[0];   // [B,L,T] f32
    const float* trans = (const float*)d_in[1];   // [T,T]   f32
    const int*   tags  = (const int*)d_in[2];     // [B,L]   i32
    const int*   lens  = (const int*)d_in[3];     // [B]     i32
    float* out  = (float*)d_out;                  // scalar mean loss
    float* fwd  = (float*)d_ws;                   // [B] forward scores
    float* gold = fwd + BB;                       // [B] gold scores

    crf_forward<<<BB / NB, 256, 0, stream>>>(feats, trans, lens, fwd);
    crf_gold<<<BB, 256, 0, stream>>>(feats, trans, tags, lens, gold);
    crf_final<<<1, 256, 0, stream>>>(fwd, gold, out);
}